// Column_72722386255974
// MI455X (gfx1250) — compile-verified
//
#include <hip/hip_runtime.h>
#include <hip/hip_bf16.h>

typedef __attribute__((ext_vector_type(2))) float v2f;
typedef __attribute__((ext_vector_type(4))) float v4f;
typedef __attribute__((ext_vector_type(8))) float v8f;

#define TDIM   32
#define NFEAT  512
#define KDIM   262144        // C*RF*L = 1*64*4096
#define THRESH 0.2f
#define KWTA   8

#define KSPLIT 16            // blocks along K
#define WAVES  8             // waves per block
#define KPERWAVE (KDIM / KSPLIT / WAVES)   // 2048 -> 256 8-K steps

__global__ void zero_f32(float* __restrict__ p, int n) {
    int i = blockIdx.x * blockDim.x + threadIdx.x;
    if (i < n) p[i] = 0.0f;
}

// Split-K GEMM: acc[t, o] += sum_k A[t,k] * W[o,k]   (fp32 WMMA 16x16x4)
// K mapping inside each 8-K step is permuted so each lane loads 4 contiguous
// floats (one B128) shared identically by A and B fragments — sum over K is
// order-invariant, so the result is unchanged.
__global__ void __launch_bounds__(WAVES * 32)
colgemm_wmma(const float* __restrict__ A,   // rec_field: [32, 262144] row-major
             const float* __restrict__ W,   // weight:    [512, 262144] row-major
             float* __restrict__ acc)       // [32, 512] fp32 accumulator
{
    const int lane  = threadIdx.x & 31;
    const int wave  = threadIdx.x >> 5;
    const int ntile = blockIdx.x;                       // 0..31  (N tile of 16)
    const int kblk  = blockIdx.y;                       // 0..KSPLIT-1

    const size_t k0 = (size_t)kblk * (KDIM / KSPLIT) + (size_t)wave * KPERWAVE;

    const int row = lane & 15;                          // M-row (A) / N-col (B)
    const int g4  = (lane >> 4) * 4;                    // lanes 16-31 own k+4..k+7

    const float* pa0 = A + (size_t)row * KDIM        + k0 + g4;  // t = row
    const float* pa1 = A + (size_t)(row + 16) * KDIM + k0 + g4;  // t = row+16
    const float* pb  = W + (size_t)(ntile * 16 + row) * KDIM + k0 + g4;

    v8f c0 = {};   // D for t in [0,16)
    v8f c1 = {};   // D for t in [16,32)

    constexpr int STEPS = KPERWAVE / 8;                 // 256, compile-time
    #pragma unroll 4
    for (int i = 0; i < STEPS; ++i) {
        const size_t off = (size_t)i * 8;
        v4f av0 = *(const v4f*)(pa0 + off);
        v4f av1 = *(const v4f*)(pa1 + off);
        v4f bv  = *(const v4f*)(pb  + off);

        v2f a0lo = {av0.x, av0.y}, a0hi = {av0.z, av0.w};
        v2f a1lo = {av1.x, av1.y}, a1hi = {av1.z, av1.w};
        v2f blo  = {bv.x,  bv.y },  bhi = {bv.z,  bv.w };

        // 8 args: (neg_a, A, neg_b, B, c_mod, C, reuse_a, reuse_b)
        c0 = __builtin_amdgcn_wmma_f32_16x16x4_f32(false, a0lo, false, blo,
                                                   (short)0, c0, false, false);
        c1 = __builtin_amdgcn_wmma_f32_16x16x4_f32(false, a1lo, false, blo,
                                                   (short)0, c1, false, false);
        c0 = __builtin_amdgcn_wmma_f32_16x16x4_f32(false, a0hi, false, bhi,
                                                   (short)0, c0, false, false);
        c1 = __builtin_amdgcn_wmma_f32_16x16x4_f32(false, a1hi, false, bhi,
                                                   (short)0, c1, false, false);
    }

    // C/D layout: VGPR i -> M = i (lanes 0-15) or M = 8+i (lanes 16-31); N = lane&15
    const int n   = ntile * 16 + row;
    const int mhi = (lane >> 4) * 8;
    #pragma unroll
    for (int i = 0; i < 8; ++i) {
        const int t0 = mhi + i;
        atomicAdd(&acc[(size_t)t0 * NFEAT + n],        c0[i]);
        atomicAdd(&acc[(size_t)(t0 + 16) * NFEAT + n], c1[i]);
    }
}

// Threshold + first-spike + iterative k-WTA + spike output. One block, 512 threads.
__global__ void __launch_bounds__(NFEAT)
kwta_fire(const float* __restrict__ acc, float* __restrict__ out)
{
    __shared__ float sval[NFEAT];
    __shared__ int   sidx[NFEAT];
    __shared__ float totals[NFEAT];
    __shared__ unsigned char sel[NFEAT];

    const int o = threadIdx.x;

    // per-feature thresholded potentials + spike stats
    float pots[TDIM];
    int nspk = 0;
    #pragma unroll
    for (int t = 0; t < TDIM; ++t) {
        float p = acc[(size_t)t * NFEAT + o];
        p = (p < THRESH) ? 0.0f : p;
        pots[t] = p;
        nspk += (p > 0.0f) ? 1 : 0;
    }
    int first = TDIM - nspk;                 // earliest spike index (clipped)
    if (first > TDIM - 1) first = TDIM - 1;
    if (first < 0) first = 0;
    const float value = pots[first];

    // v = max(trunc) * T ; trunc max = max over spiking features of `value`
    sval[o] = (nspk > 0) ? value : 0.0f;
    __syncthreads();
    for (int s = NFEAT / 2; s > 0; s >>= 1) {
        if (o < s) sval[o] = fmaxf(sval[o], sval[o + s]);
        __syncthreads();
    }
    const float v = sval[0] * (float)TDIM;
    __syncthreads();

    totals[o] = (float)nspk * (value + v);
    sel[o] = 0;
    __syncthreads();

    // KWTA argmax iterations (ties -> lowest index, matching jnp.argmax)
    for (int it = 0; it < KWTA; ++it) {
        sval[o] = totals[o];
        sidx[o] = o;
        __syncthreads();
        for (int s = NFEAT / 2; s > 0; s >>= 1) {
            if (o < s) {
                const float v1 = sval[o], v2 = sval[o + s];
                const int   i1 = sidx[o], i2 = sidx[o + s];
                if (v2 > v1 || (v2 == v1 && i2 < i1)) { sval[o] = v2; sidx[o] = i2; }
            }
            __syncthreads();
        }
        if (o == 0) {
            const int w = sidx[0];
            if (sval[0] != 0.0f) sel[w] = 1;
            totals[w] = 0.0f;               // inhibit winner
        }
        __syncthreads();
    }

    const float s = sel[o] ? 1.0f : 0.0f;
    #pragma unroll
    for (int t = 0; t < TDIM; ++t)
        out[(size_t)t * NFEAT + o] = (pots[t] > 0.0f) ? s : 0.0f;
}

extern "C" void kernel_launch(void* const* d_in, const int* in_sizes, int n_in,
                              void* d_out, int out_size, void* d_ws, size_t ws_size,
                              hipStream_t stream) {
    const float* rec = (const float*)d_in[0];   // (32, 1, 64, 4096) fp32
    const float* wgt = (const float*)d_in[1];   // (512, 1, 64, 4096) fp32
    float* out = (float*)d_out;                 // (32, 512, 1, 1) fp32
    float* acc = (float*)d_ws;                  // 32*512 fp32 accumulator

    const int accN = TDIM * NFEAT;
    zero_f32<<<(accN + 255) / 256, 256, 0, stream>>>(acc, accN);

    dim3 grid(NFEAT / 16, KSPLIT);              // 32 N-tiles x 16 K splits
    colgemm_wmma<<<grid, WAVES * 32, 0, stream>>>(rec, wgt, acc);

    kwta_fire<<<1, NFEAT, 0, stream>>>(acc, out);
}